// spGNN_10256381903669
// MI455X (gfx1250) — compile-verified
//
#include <hip/hip_runtime.h>
#include <math.h>

typedef float v2f __attribute__((ext_vector_type(2)));
typedef float v8f __attribute__((ext_vector_type(8)));

#define F_IN  96
#define F_HID 384
#define F_OUT 192
#define BN_EPS 1e-5f

__device__ __forceinline__ float elu1(float x) {
    return x > 0.0f ? x : (expf(x) - 1.0f);
}

// ---------------------------------------------------------------------------
// 1) v = epsilon * x   (self-loop term), and zero the BN stat accumulators
// ---------------------------------------------------------------------------
__global__ void k_init_v(const float* __restrict__ x, const float* __restrict__ eps_p,
                         float* __restrict__ v, float* __restrict__ stats,
                         int totalV, int statsN) {
    int i = blockIdx.x * blockDim.x + threadIdx.x;
    float eps = eps_p[0];
    if (i < totalV) v[i] = eps * x[i];
    if (i < statsN) stats[i] = 0.0f;
}

// ---------------------------------------------------------------------------
// 2) SpMM scatter: v[row[e]] += val[e] * x[col[e]].  x (19MB) is L2-resident;
//    24 threads per edge, 4 features each -> coalesced gather + f32 atomics.
// ---------------------------------------------------------------------------
__global__ void k_spmm(const float* __restrict__ x, const float* __restrict__ val,
                       const int* __restrict__ row, const int* __restrict__ col,
                       float* __restrict__ v, int E) {
    int idx = blockIdx.x * blockDim.x + threadIdx.x;
    int e = idx / (F_IN / 4);
    if (e >= E) return;
    int f = (idx % (F_IN / 4)) * 4;
    int c = col[e];
    int r = row[e];
    float w = val[e];
    const float4 xv = *(const float4*)(x + (size_t)c * F_IN + f);
    float* dst = v + (size_t)r * F_IN + f;
    atomicAdd(dst + 0, w * xv.x);
    atomicAdd(dst + 1, w * xv.y);
    atomicAdd(dst + 2, w * xv.z);
    atomicAdd(dst + 3, w * xv.w);
}

// ---------------------------------------------------------------------------
// 3) GEMM1: t1[N,384] = v[N,96] @ W1[96,384] + b1   (fp32 WMMA 16x16x4)
//    6 waves/block; each wave register-blocks 4 N-tiles (64 cols), reusing
//    its A fragment across 4 WMMAs per K-step.
// ---------------------------------------------------------------------------
__global__ void k_gemm1(const float* __restrict__ V, const float* __restrict__ W1,
                        const float* __restrict__ b1, float* __restrict__ T1, int Nn) {
    const int wave = threadIdx.x >> 5;        // 0..5 -> 64-col group
    const int lane = threadIdx.x & 31;
    const int half = lane >> 4;
    const int l    = lane & 15;
    const int m0 = blockIdx.x * 16;
    const int n0 = wave * 64;
    int m = m0 + l;
    if (m >= Nn) m = Nn - 1;                  // clamp loads (ragged tail)
    const float* arow = V + (size_t)m * F_IN;

    v8f acc0 = {}, acc1 = {}, acc2 = {}, acc3 = {};
    for (int k = 0; k < F_IN; k += 4) {
        const int ka = k + 2 * half;          // A: VGPR0=K{0,2}, VGPR1=K{1,3}
        const float2 av = *(const float2*)(arow + ka);
        v2f a;  a.x = av.x;  a.y = av.y;
        const float* w0 = W1 + (size_t)ka * F_HID + n0 + l;
        const float* w1 = W1 + (size_t)(ka + 1) * F_HID + n0 + l;
        v2f b;
        b.x = w0[0];   b.y = w1[0];
        acc0 = __builtin_amdgcn_wmma_f32_16x16x4_f32(false, a, false, b, (short)0, acc0, false, false);
        b.x = w0[16];  b.y = w1[16];
        acc1 = __builtin_amdgcn_wmma_f32_16x16x4_f32(false, a, false, b, (short)0, acc1, false, false);
        b.x = w0[32];  b.y = w1[32];
        acc2 = __builtin_amdgcn_wmma_f32_16x16x4_f32(false, a, false, b, (short)0, acc2, false, false);
        b.x = w0[48];  b.y = w1[48];
        acc3 = __builtin_amdgcn_wmma_f32_16x16x4_f32(false, a, false, b, (short)0, acc3, false, false);
    }
    const float bi0 = b1[n0 + l];
    const float bi1 = b1[n0 + 16 + l];
    const float bi2 = b1[n0 + 32 + l];
    const float bi3 = b1[n0 + 48 + l];
    float* base = T1 + (size_t)(m0 + 8 * half) * F_HID + n0 + l;
    if (m0 + 16 <= Nn) {                      // full-tile fast path: no guards
        #pragma unroll
        for (int r = 0; r < 8; ++r) {
            float* rp = base + (size_t)r * F_HID;
            rp[0]  = acc0[r] + bi0;
            rp[16] = acc1[r] + bi1;
            rp[32] = acc2[r] + bi2;
            rp[48] = acc3[r] + bi3;
        }
    } else {
        for (int r = 0; r < 8; ++r) {
            if (m0 + 8 * half + r < Nn) {
                float* rp = base + (size_t)r * F_HID;
                rp[0]  = acc0[r] + bi0;
                rp[16] = acc1[r] + bi1;
                rp[32] = acc2[r] + bi2;
                rp[48] = acc3[r] + bi3;
            }
        }
    }
}

// ---------------------------------------------------------------------------
// 4) Per-feature column sum / sum-of-squares (coalesced row-chunk partials)
// ---------------------------------------------------------------------------
__global__ void k_colstats(const float* __restrict__ M, int Nrows, int F,
                           float* __restrict__ sum, float* __restrict__ sq) {
    const int r0 = blockIdx.x * 128;
    const int rend = min(r0 + 128, Nrows);
    for (int f = threadIdx.x; f < F; f += blockDim.x) {
        float s = 0.0f, ss = 0.0f;
        for (int r = r0; r < rend; ++r) {
            float t = M[(size_t)r * F + f];
            s += t;
            ss += t * t;
        }
        atomicAdd(&sum[f], s);
        atomicAdd(&sq[f], ss);
    }
}

// ---------------------------------------------------------------------------
// 5) BN finalize: scale = g * rsqrt(var + eps); shift = beta - mean*scale
// ---------------------------------------------------------------------------
__global__ void k_bn_finalize(const float* __restrict__ sum, const float* __restrict__ sq,
                              const float* __restrict__ g, const float* __restrict__ be,
                              float invN, int F,
                              float* __restrict__ sc, float* __restrict__ sh) {
    int f = blockIdx.x * blockDim.x + threadIdx.x;
    if (f >= F) return;
    float mean = sum[f] * invN;
    float var  = sq[f] * invN - mean * mean;   // biased variance
    float s = g[f] * rsqrtf(var + BN_EPS);
    sc[f] = s;
    sh[f] = be[f] - mean * s;
}

// ---------------------------------------------------------------------------
// 6) Elementwise BN + ELU, one row per block, float4 lanes (no modulo)
// ---------------------------------------------------------------------------
__global__ void k_bnelu_rows(const float* __restrict__ in, float* __restrict__ out,
                             const float* __restrict__ sc, const float* __restrict__ sh,
                             int F) {
    const int f4 = threadIdx.x * 4;
    if (f4 >= F) return;
    const size_t base = (size_t)blockIdx.x * F + f4;
    const float4 vv = *(const float4*)(in + base);
    const float4 s  = *(const float4*)(sc + f4);
    const float4 t  = *(const float4*)(sh + f4);
    float4 o;
    o.x = elu1(vv.x * s.x + t.x);
    o.y = elu1(vv.y * s.y + t.y);
    o.z = elu1(vv.z * s.z + t.z);
    o.w = elu1(vv.w * s.w + t.w);
    *(float4*)(out + base) = o;
}

// ---------------------------------------------------------------------------
// 7) GEMM2: t2[N,192] = h[N,384] @ W2[384,192] + b2   (fp32 WMMA 16x16x4)
//    12 waves/block = 4 M-strips x 3 waves; each wave does 4 N-tiles.
// ---------------------------------------------------------------------------
__global__ void k_gemm2(const float* __restrict__ H, const float* __restrict__ W2,
                        const float* __restrict__ b2, float* __restrict__ T2, int Nn) {
    const int wave  = threadIdx.x >> 5;       // 0..11
    const int strip = wave / 3;               // 0..3 -> M strip in block
    const int wn    = wave % 3;               // 0..2 -> 64-col group
    const int lane = threadIdx.x & 31;
    const int half = lane >> 4;
    const int l    = lane & 15;
    const int m0 = blockIdx.x * 64 + strip * 16;
    const int n0 = wn * 64;
    if (m0 >= Nn) return;                     // whole-wave skip on ragged grid
    int m = m0 + l;
    if (m >= Nn) m = Nn - 1;
    const float* arow = H + (size_t)m * F_HID;

    v8f acc0 = {}, acc1 = {}, acc2 = {}, acc3 = {};
    for (int k = 0; k < F_HID; k += 4) {
        const int ka = k + 2 * half;
        const float2 av = *(const float2*)(arow + ka);
        v2f a;  a.x = av.x;  a.y = av.y;
        const float* w0 = W2 + (size_t)ka * F_OUT + n0 + l;
        const float* w1 = W2 + (size_t)(ka + 1) * F_OUT + n0 + l;
        v2f b;
        b.x = w0[0];   b.y = w1[0];
        acc0 = __builtin_amdgcn_wmma_f32_16x16x4_f32(false, a, false, b, (short)0, acc0, false, false);
        b.x = w0[16];  b.y = w1[16];
        acc1 = __builtin_amdgcn_wmma_f32_16x16x4_f32(false, a, false, b, (short)0, acc1, false, false);
        b.x = w0[32];  b.y = w1[32];
        acc2 = __builtin_amdgcn_wmma_f32_16x16x4_f32(false, a, false, b, (short)0, acc2, false, false);
        b.x = w0[48];  b.y = w1[48];
        acc3 = __builtin_amdgcn_wmma_f32_16x16x4_f32(false, a, false, b, (short)0, acc3, false, false);
    }
    const float bi0 = b2[n0 + l];
    const float bi1 = b2[n0 + 16 + l];
    const float bi2 = b2[n0 + 32 + l];
    const float bi3 = b2[n0 + 48 + l];
    float* base = T2 + (size_t)(m0 + 8 * half) * F_OUT + n0 + l;
    if (m0 + 16 <= Nn) {
        #pragma unroll
        for (int r = 0; r < 8; ++r) {
            float* rp = base + (size_t)r * F_OUT;
            rp[0]  = acc0[r] + bi0;
            rp[16] = acc1[r] + bi1;
            rp[32] = acc2[r] + bi2;
            rp[48] = acc3[r] + bi3;
        }
    } else {
        for (int r = 0; r < 8; ++r) {
            if (m0 + 8 * half + r < Nn) {
                float* rp = base + (size_t)r * F_OUT;
                rp[0]  = acc0[r] + bi0;
                rp[16] = acc1[r] + bi1;
                rp[32] = acc2[r] + bi2;
                rp[48] = acc3[r] + bi3;
            }
        }
    }
}

// ---------------------------------------------------------------------------
extern "C" void kernel_launch(void* const* d_in, const int* in_sizes, int n_in,
                              void* d_out, int out_size, void* d_ws, size_t ws_size,
                              hipStream_t stream) {
    const float* x    = (const float*)d_in[0];
    const float* val  = (const float*)d_in[1];
    const float* W1   = (const float*)d_in[2];
    const float* b1   = (const float*)d_in[3];
    const float* g1   = (const float*)d_in[4];
    const float* be1  = (const float*)d_in[5];
    const float* W2   = (const float*)d_in[6];
    const float* b2   = (const float*)d_in[7];
    const float* g2   = (const float*)d_in[8];
    const float* be2  = (const float*)d_in[9];
    const float* eps  = (const float*)d_in[10];
    const int*   row  = (const int*)d_in[11];
    const int*   col  = (const int*)d_in[12];
    float* out = (float*)d_out;

    const int N = in_sizes[0] / F_IN;   // 50000
    const int E = in_sizes[1];          // 800000

    // Workspace layout (floats)
    float* ws   = (float*)d_ws;
    float* v    = ws;                                   // N*96
    float* t1   = v  + (size_t)N * F_IN;                // N*384
    float* t2   = t1 + (size_t)N * F_HID;               // N*192
    float* st   = t2 + (size_t)N * F_OUT;               // stats block
    float* sum1 = st;            float* sq1 = st + 384;
    float* sc1  = st + 768;      float* sh1 = st + 1152;
    float* sum2 = st + 1536;     float* sq2 = st + 1728;
    float* sc2  = st + 1920;     float* sh2 = st + 2112;
    const int statsN = 2304;

    const float invN = 1.0f / (float)N;
    const int mtiles = (N + 15) / 16;

    // 1) v = eps*x, zero stats
    {
        int total = N * F_IN;
        int blocks = (total + 255) / 256;
        k_init_v<<<blocks, 256, 0, stream>>>(x, eps, v, st, total, statsN);
    }
    // 2) edge scatter (L2 atomics)
    {
        long long work = (long long)E * (F_IN / 4);
        int blocks = (int)((work + 255) / 256);
        k_spmm<<<blocks, 256, 0, stream>>>(x, val, row, col, v, E);
    }
    // 3) GEMM1 (WMMA f32): t1 = v @ W1 + b1
    k_gemm1<<<mtiles, 192, 0, stream>>>(v, W1, b1, t1, N);
    // 4-5) BN1 statistics + finalize
    k_colstats<<<(N + 127) / 128, 256, 0, stream>>>(t1, N, F_HID, sum1, sq1);
    k_bn_finalize<<<(F_HID + 255) / 256, 256, 0, stream>>>(sum1, sq1, g1, be1, invN, F_HID, sc1, sh1);
    // 6) h = elu(bn(t1)) in place, float4, one row per block
    k_bnelu_rows<<<N, F_HID / 4, 0, stream>>>(t1, t1, sc1, sh1, F_HID);
    // 7) GEMM2 (WMMA f32): t2 = h @ W2 + b2
    k_gemm2<<<(mtiles + 3) / 4, 384, 0, stream>>>(t1, W2, b2, t2, N);
    // 8-9) BN2 statistics + finalize
    k_colstats<<<(N + 127) / 128, 256, 0, stream>>>(t2, N, F_OUT, sum2, sq2);
    k_bn_finalize<<<(F_OUT + 255) / 256, 256, 0, stream>>>(sum2, sq2, g2, be2, invN, F_OUT, sc2, sh2);
    // 10) out = elu(bn(t2))
    k_bnelu_rows<<<N, F_OUT / 4, 0, stream>>>(t2, out, sc2, sh2, F_OUT);
}